// FlowLayer_19885698581167
// MI455X (gfx1250) — compile-verified
//
#include <hip/hip_runtime.h>
#include <math.h>

// ---------------- problem constants ----------------
#define PN     8          // batch N
#define PC     64         // channels C
#define PH     16
#define PW     16
#define NPIX   256        // H*W
#define EQN    10
#define GR     4
#define NGRP   (PN*GR)    // 32 systems
#define MATN   1024       // 4*num
#define NROWS  (NPIX*EQN) // 2560 equations per system
#define NB     16         // Cholesky panel width
#define NSTEPS (MATN/NB)  // 64
#define CHUNK  8          // matrices factored concurrently (32MB scratch)

typedef __attribute__((ext_vector_type(2))) float v2f;
typedef __attribute__((ext_vector_type(8))) float v8f;

__device__ __forceinline__ float lrelu(float v) { return v >= 0.f ? v : 0.01f * v; }
__device__ __forceinline__ float sigm(float v)  { return 1.0f / (1.0f + __expf(-v)); }

// ---------------- 1) extract sparse structure of `a` (<=2 nnz/row) --------
__global__ void k_sparsify_a(const float* __restrict__ a,
                             int* __restrict__ rowc, float* __restrict__ rowv) {
  int r = blockIdx.x * blockDim.x + threadIdx.x;
  if (r >= NROWS) return;
  int c0 = -1, c1 = -1; float v0 = 0.f, v1 = 0.f;
  const float* ap = a + (size_t)r * MATN;
  for (int j = 0; j < MATN; ++j) {
    float x = ap[j];
    if (x != 0.f) {
      if (c0 < 0) { c0 = j; v0 = x; }
      else if (c1 < 0) { c1 = j; v1 = x; }
    }
  }
  rowc[2*r] = c0; rowc[2*r+1] = c1;
  rowv[2*r] = v0; rowv[2*r+1] = v1;
}

// ---------------- 2) generic 3x3 SAME conv on 16x16 maps -------------------
// act: 0=none, 1=lrelu, 2=sigmoid
__global__ void k_conv3x3(const float* __restrict__ in, const float* __restrict__ wgt,
                          const float* __restrict__ bias, float* __restrict__ out,
                          int n_, int cin, int cout, int act) {
  int idx = blockIdx.x * blockDim.x + threadIdx.x;
  int total = n_ * cout * NPIX;
  if (idx >= total) return;
  int pix = idx & 255;
  int oc  = (idx >> 8) % cout;
  int n   = idx / (cout * NPIX);
  int h = pix >> 4, w = pix & 15;
  float acc = bias[oc];
  for (int ic = 0; ic < cin; ++ic) {
    const float* ip = in + ((size_t)n * cin + ic) * NPIX;
    const float* wp = wgt + ((size_t)oc * cin + ic) * 9;
    __builtin_prefetch(ip, 0, 1);     // global_prefetch_b8
    #pragma unroll
    for (int ky = -1; ky <= 1; ++ky) {
      int hh = h + ky;
      if ((unsigned)hh >= 16u) continue;
      #pragma unroll
      for (int kx = -1; kx <= 1; ++kx) {
        int ww = w + kx;
        if ((unsigned)ww >= 16u) continue;
        acc = fmaf(ip[hh*16 + ww], wp[(ky+1)*3 + (kx+1)], acc);
      }
    }
  }
  if (act == 1) acc = lrelu(acc);
  else if (act == 2) acc = sigm(acc);
  out[idx] = acc;
}

// ---------------- 3) SE branch: pool -> 1x1 -> lrelu -> 1x1 -> sigmoid -----
__global__ void k_se(const float* __restrict__ x,
                     const float* __restrict__ sw1, const float* __restrict__ sb1,
                     const float* __restrict__ sw2, const float* __restrict__ sb2,
                     float* __restrict__ se) {
  int n = blockIdx.x, tid = threadIdx.x;           // 64 threads
  __shared__ float pooled[PC];
  __shared__ float h1[PC/2];
  const float* xp = x + ((size_t)n * PC + tid) * NPIX;
  float s = 0.f;
  for (int p = 0; p < NPIX; ++p) s += xp[p];
  pooled[tid] = s * (1.0f / NPIX);
  __syncthreads();
  if (tid < PC/2) {
    float acc = sb1[tid];
    for (int ic = 0; ic < PC; ++ic) acc = fmaf(pooled[ic], sw1[tid*PC + ic], acc);
    h1[tid] = lrelu(acc);
  }
  __syncthreads();
  if (tid < 4*GR) {
    float acc = sb2[tid];
    for (int ic = 0; ic < PC/2; ++ic) acc = fmaf(h1[ic], sw2[tid*(PC/2) + ic], acc);
    se[n*(4*GR) + tid] = sigm(acc);
  }
}

// ---------------- 4a) init ATA chunk = eps*I, zero ATB chunk ---------------
__global__ void k_fill(float* __restrict__ ATA, float* __restrict__ ATBc) {
  size_t total = (size_t)CHUNK * MATN * MATN;
  for (size_t i = (size_t)blockIdx.x * blockDim.x + threadIdx.x;
       i < total; i += (size_t)gridDim.x * blockDim.x) {
    unsigned within = (unsigned)(i & (MATN*MATN - 1));
    unsigned r = within >> 10, c = within & (MATN - 1);
    ATA[i] = (r == c) ? 1e-12f : 0.0f;
    if (i < (size_t)CHUNK * MATN) ATBc[i] = 0.0f;
  }
}

// ---------------- 4b) scatter-assemble ATA (graph Laplacian) + ATB ---------
__global__ void k_build(const float* __restrict__ att, const float* __restrict__ g,
                        const int* __restrict__ rowc, const float* __restrict__ rowv,
                        float* __restrict__ ATA, float* __restrict__ ATBc, int batch0) {
  int lb = blockIdx.y;                                // local matrix in chunk
  int r  = blockIdx.x * blockDim.x + threadIdx.x;     // equation row
  if (r >= NROWS) return;
  int pix = r / EQN, e = r % EQN;
  int b = batch0 + lb, n = b >> 2, grp = b & 3;
  int ch = grp * EQN + e;
  float w  = att[((size_t)n * (EQN*GR) + ch) * NPIX + pix];
  float gv = g  [((size_t)n * (EQN*GR) + ch) * NPIX + pix];
  float w2 = w * w;
  int   c0 = rowc[2*r], c1 = rowc[2*r+1];
  float v0 = rowv[2*r], v1 = rowv[2*r+1];
  if (c0 < 0) return;
  float* A = ATA + (size_t)lb * MATN * MATN;
  float* B = ATBc + (size_t)lb * MATN;
  atomicAdd(&A[(size_t)c0*MATN + c0], w2 * v0 * v0);
  atomicAdd(&B[c0], w2 * v0 * gv);
  if (c1 >= 0) {
    float cx = w2 * v0 * v1;
    atomicAdd(&A[(size_t)c0*MATN + c1], cx);
    atomicAdd(&A[(size_t)c1*MATN + c0], cx);
    atomicAdd(&A[(size_t)c1*MATN + c1], w2 * v1 * v1);
    atomicAdd(&B[c1], w2 * v1 * gv);
  }
}

// ---------------- 4c) blocked Cholesky, WMMA f32 trailing update -----------
// One workgroup (16 waves) per matrix. Panel (<=1008x16 f32 = 63KB) staged in
// LDS; trailing SYRK tiles computed 16x16 per wave with V_WMMA_F32_16X16X4_F32.
__global__ void __launch_bounds__(512)
k_cholesky(float* __restrict__ ATA) {
  float* M = ATA + (size_t)blockIdx.x * MATN * MATN;
  __shared__ float sdiag[NB*NB];
  __shared__ float spanel[(MATN-NB)*NB];   // 63KB, fits 320KB WGP LDS
  const int tid  = threadIdx.x;
  const int lane = tid & 31;
  const int wave = tid >> 5;
  const int nwaves = blockDim.x >> 5;
  const int half = lane >> 4;        // 0 for lanes 0-15, 1 for 16-31
  const int l16  = lane & 15;

  for (int k = 0; k < NSTEPS; ++k) {
    const int k0 = k * NB;
    // load diagonal block
    if (tid < NB*NB)
      sdiag[tid] = M[(size_t)(k0 + tid/NB)*MATN + k0 + (tid%NB)];
    __syncthreads();
    // tiny unblocked Cholesky of the 16x16 diagonal (serial, ~700 flops)
    if (tid == 0) {
      for (int j = 0; j < NB; ++j) {
        float d = sdiag[j*NB+j];
        for (int t = 0; t < j; ++t) d -= sdiag[j*NB+t]*sdiag[j*NB+t];
        d = sqrtf(fmaxf(d, 1e-20f));
        sdiag[j*NB+j] = d;
        float inv = 1.0f / d;
        for (int i = j+1; i < NB; ++i) {
          float s = sdiag[i*NB+j];
          for (int t = 0; t < j; ++t) s -= sdiag[i*NB+t]*sdiag[j*NB+t];
          sdiag[i*NB+j] = s * inv;
        }
      }
    }
    __syncthreads();
    if (tid < NB*NB) {
      int rr = tid / NB, cc = tid % NB;
      if (cc <= rr) M[(size_t)(k0+rr)*MATN + k0 + cc] = sdiag[tid];
    }
    const int rrem = MATN - (k0 + NB);
    // panel triangular solve: L[i,:] = A[i,:] * inv(Lkk)^T ; stage into LDS
    for (int i = tid; i < rrem; i += blockDim.x) {
      const int gi = k0 + NB + i;
      float row[NB];
      #pragma unroll
      for (int c = 0; c < NB; ++c) row[c] = M[(size_t)gi*MATN + k0 + c];
      #pragma unroll
      for (int j = 0; j < NB; ++j) {
        float s = row[j];
        for (int t = 0; t < j; ++t) s -= row[t] * sdiag[j*NB+t];
        row[j] = s / sdiag[j*NB+j];
      }
      #pragma unroll
      for (int c = 0; c < NB; ++c) {
        M[(size_t)gi*MATN + k0 + c] = row[c];
        spanel[i*NB + c] = row[c];
      }
    }
    __syncthreads();
    // trailing update: C(I,J) -= P_I * P_J^T over lower-triangular 16x16 tiles
    const int nt = rrem / NB;
    const int ntiles = nt * (nt + 1) / 2;
    for (int t = wave; t < ntiles; t += nwaves) {
      int bi = (int)floorf((sqrtf(8.0f*(float)t + 1.0f) - 1.0f) * 0.5f);
      while ((bi+1)*(bi+2)/2 <= t) ++bi;
      while (bi*(bi+1)/2 > t) --bi;
      int bj = t - bi*(bi+1)/2;
      const int gi = k0 + NB + bi*NB;
      const int gj = k0 + NB + bj*NB;
      // C/D layout: VGPR v holds M = v + 8*half, N = l16
      v8f c;
      #pragma unroll
      for (int v = 0; v < 8; ++v)
        c[v] = M[(size_t)(gi + v + 8*half)*MATN + gj + l16];
      #pragma unroll
      for (int q = 0; q < 4; ++q) {       // K = 16 split into 4x WMMA(K=4)
        v2f aop, bop;
        // A(16x4): lane holds M=l16, K = q*4 + 2*half + v ; negate for C -= A*B^T
        aop[0] = -spanel[(bi*NB + l16)*NB + q*4 + 2*half + 0];
        aop[1] = -spanel[(bi*NB + l16)*NB + q*4 + 2*half + 1];
        // B(4x16) = P_J^T : B[K,N] = P_J[N][K]
        bop[0] =  spanel[(bj*NB + l16)*NB + q*4 + 2*half + 0];
        bop[1] =  spanel[(bj*NB + l16)*NB + q*4 + 2*half + 1];
        c = __builtin_amdgcn_wmma_f32_16x16x4_f32(
              /*neg_a=*/false, aop, /*neg_b=*/false, bop,
              /*c_mod=*/(short)0, c, /*reuse_a=*/false, /*reuse_b=*/false);
      }
      #pragma unroll
      for (int v = 0; v < 8; ++v)
        M[(size_t)(gi + v + 8*half)*MATN + gj + l16] = c[v];
    }
    __syncthreads();
  }
}

// ---------------- 4d) blocked L y = b ; L^T x = y --------------------------
__global__ void k_trsv(const float* __restrict__ ATA, const float* __restrict__ ATBc,
                       float* __restrict__ sol, int batch0) {
  const float* M = ATA + (size_t)blockIdx.x * MATN * MATN;
  const int tid = threadIdx.x;
  __shared__ float sx[MATN];
  for (int i = tid; i < MATN; i += blockDim.x)
    sx[i] = ATBc[(size_t)blockIdx.x * MATN + i];
  __syncthreads();
  // forward
  for (int kb = 0; kb < NSTEPS; ++kb) {
    int k0 = kb * NB;
    if (tid == 0) {
      for (int j = k0; j < k0 + NB; ++j) {
        float s = sx[j];
        for (int t = k0; t < j; ++t) s -= M[(size_t)j*MATN + t] * sx[t];
        sx[j] = s / M[(size_t)j*MATN + j];
      }
    }
    __syncthreads();
    for (int i = k0 + NB + tid; i < MATN; i += blockDim.x) {
      float s = sx[i];
      #pragma unroll
      for (int t = 0; t < NB; ++t) s -= M[(size_t)i*MATN + k0 + t] * sx[k0 + t];
      sx[i] = s;
    }
    __syncthreads();
  }
  // backward (L^T)
  for (int kb = NSTEPS - 1; kb >= 0; --kb) {
    int k0 = kb * NB;
    if (tid == 0) {
      for (int j = k0 + NB - 1; j >= k0; --j) {
        float s = sx[j];
        for (int t = j + 1; t < k0 + NB; ++t) s -= M[(size_t)t*MATN + j] * sx[t];
        sx[j] = s / M[(size_t)j*MATN + j];
      }
    }
    __syncthreads();
    for (int i = tid; i < k0; i += blockDim.x) {
      float s = sx[i];
      #pragma unroll
      for (int t = 0; t < NB; ++t) s -= M[(size_t)(k0 + t)*MATN + i] * sx[k0 + t];
      sx[i] = s;
    }
    __syncthreads();
  }
  for (int i = tid; i < MATN; i += blockDim.x)
    sol[(size_t)(batch0 + blockIdx.x) * MATN + i] = sx[i];
}

// ---------------- 5) GroupNorm(1) + affine + SE scale ----------------------
__global__ void k_gnorm_se(const float* __restrict__ sol, const float* __restrict__ se,
                           const float* __restrict__ insw, const float* __restrict__ insb,
                           float* __restrict__ y) {
  int n = blockIdx.x, tid = threadIdx.x;   // 256 threads per sample
  __shared__ float r1[256], r2[256];
  float s = 0.f, s2 = 0.f;
  for (int idx = tid; idx < GR*MATN; idx += 256) {
    int g = idx >> 10, rem = idx & (MATN - 1);
    float v = sol[(size_t)(n*GR + g)*MATN + rem];
    s += v; s2 += v * v;
  }
  r1[tid] = s; r2[tid] = s2;
  __syncthreads();
  for (int off = 128; off > 0; off >>= 1) {
    if (tid < off) { r1[tid] += r1[tid+off]; r2[tid] += r2[tid+off]; }
    __syncthreads();
  }
  float mu  = r1[0] * (1.0f / (GR*MATN));
  float var = r2[0] * (1.0f / (GR*MATN)) - mu*mu;
  float rstd = rsqrtf(var + 1e-5f);
  for (int idx = tid; idx < GR*MATN; idx += 256) {
    int g = idx >> 10, rem = idx & (MATN - 1);
    int d = rem >> 8, pix = rem & 255;
    int ch = g*4 + d;
    float v = sol[(size_t)(n*GR + g)*MATN + rem];
    v = (v - mu) * rstd;
    v = v * insw[ch] + insb[ch];
    v *= se[n*(4*GR) + ch];
    y[((size_t)n*(4*GR) + ch)*NPIX + pix] = v;
  }
}

// ---------------- host orchestration ---------------------------------------
extern "C" void kernel_launch(void* const* d_in, const int* in_sizes, int n_in,
                              void* d_out, int out_size, void* d_ws, size_t ws_size,
                              hipStream_t stream) {
  (void)in_sizes; (void)n_in; (void)out_size; (void)ws_size;
  const float* x    = (const float*)d_in[0];
  const float* a    = (const float*)d_in[1];
  const float* gw1  = (const float*)d_in[2];
  const float* gb1  = (const float*)d_in[3];
  const float* gw2  = (const float*)d_in[4];
  const float* gb2  = (const float*)d_in[5];
  const float* aw1  = (const float*)d_in[6];
  const float* ab1  = (const float*)d_in[7];
  const float* aw2  = (const float*)d_in[8];
  const float* ab2  = (const float*)d_in[9];
  const float* sw1  = (const float*)d_in[10];
  const float* sb1  = (const float*)d_in[11];
  const float* sw2  = (const float*)d_in[12];
  const float* sb2  = (const float*)d_in[13];
  const float* insw = (const float*)d_in[14];
  const float* insb = (const float*)d_in[15];
  const float* pw   = (const float*)d_in[16];
  const float* pb   = (const float*)d_in[17];
  float* out = (float*)d_out;

  // workspace carve-up (floats)
  float* p    = (float*)d_ws;
  float* t1   = p;                 p += PN*PC*NPIX;        // 131072
  float* att  = p;                 p += PN*EQN*GR*NPIX;    // 81920
  float* gbuf = p;                 p += PN*EQN*GR*NPIX;    // 81920
  float* sebuf= p;                 p += PN*4*GR;           // 128
  int*   rowc = (int*)p;           p += 2*NROWS;           // 5120
  float* rowv = p;                 p += 2*NROWS;           // 5120
  float* ATB  = p;                 p += NGRP*MATN;         // 32768
  float* sol  = p;                 p += NGRP*MATN;         // 32768
  float* ybuf = p;                 p += PN*4*GR*NPIX;      // 32768
  float* ATA  = p;                 // CHUNK*1024*1024 = 32MB

  // 1) structure of `a`
  k_sparsify_a<<<(NROWS+255)/256, 256, 0, stream>>>(a, rowc, rowv);
  // 2) attention branch: x -> lrelu conv -> sigmoid conv
  k_conv3x3<<<(PN*PC*NPIX+255)/256, 256, 0, stream>>>(x, aw1, ab1, t1, PN, PC, PC, 1);
  k_conv3x3<<<(PN*EQN*GR*NPIX+255)/256, 256, 0, stream>>>(t1, aw2, ab2, att, PN, PC, EQN*GR, 2);
  //    gradient branch: x -> lrelu conv -> conv
  k_conv3x3<<<(PN*PC*NPIX+255)/256, 256, 0, stream>>>(x, gw1, gb1, t1, PN, PC, PC, 1);
  k_conv3x3<<<(PN*EQN*GR*NPIX+255)/256, 256, 0, stream>>>(t1, gw2, gb2, gbuf, PN, PC, EQN*GR, 0);
  // 3) SE branch
  k_se<<<PN, PC, 0, stream>>>(x, sw1, sb1, sw2, sb2, sebuf);
  // 4) chunks of CHUNK systems: assemble -> Cholesky(WMMA) -> triangular solves
  for (int batch0 = 0; batch0 < NGRP; batch0 += CHUNK) {
    float* ATBc = ATB + (size_t)batch0 * MATN;
    k_fill<<<4096, 256, 0, stream>>>(ATA, ATBc);
    k_build<<<dim3((NROWS+255)/256, CHUNK), 256, 0, stream>>>(att, gbuf, rowc, rowv,
                                                              ATA, ATBc, batch0);
    k_cholesky<<<CHUNK, 512, 0, stream>>>(ATA);
    k_trsv<<<CHUNK, 256, 0, stream>>>(ATA, ATBc, sol, batch0);
  }
  // 5) GroupNorm + affine + SE
  k_gnorm_se<<<PN, 256, 0, stream>>>(sol, sebuf, insw, insb, ybuf);
  // 6) final 3x3 conv -> d_out (8,128,16,16)
  k_conv3x3<<<(PN*128*NPIX+255)/256, 256, 0, stream>>>(ybuf, pw, pb, out, PN, 4*GR, 128, 0);
}